// CuGraphGATConv_5282809774194
// MI455X (gfx1250) — compile-verified
//
#include <hip/hip_runtime.h>
#include <hip/hip_bf16.h>
#include <math.h>

#define N_NODES   50000
#define N_EDGES   800000
#define IN_FEATS  128
#define NUM_HEADS 8
#define OUT_FEATS 16
#define HD        (NUM_HEADS * OUT_FEATS)   // 128
#define NEG_SLOPE 0.2f

typedef float v2f __attribute__((ext_vector_type(2)));
typedef float v8f __attribute__((ext_vector_type(8)));

// ---------------------------------------------------------------------------
// Kernel 1: h = feat @ W using fp32 WMMA (V_WMMA_F32_16X16X4_F32).
// Grid: 3125 blocks (one 16-row block of feat each), 256 threads = 8 waves,
// wave w computes output N-tile [w*16, w*16+16). A-tile staged in LDS once.
// ---------------------------------------------------------------------------
__global__ __launch_bounds__(256)
void gat_gemm_wmma(const float* __restrict__ feat,
                   const float* __restrict__ W,
                   float* __restrict__ h) {
  __shared__ float lA[16 * IN_FEATS];   // 8 KB, 16 rows x 128 K
  const int tid  = threadIdx.x;
  const int wave = tid >> 5;
  const int lane = tid & 31;
  const int m0   = blockIdx.x * 16;
  const int n0   = wave * 16;

  // Cooperative load of A tile (16 x 128 fp32) into LDS.
  #pragma unroll
  for (int i = tid; i < 16 * IN_FEATS; i += 256) {
    const int r = i >> 7;       // / 128
    const int c = i & 127;
    lA[i] = feat[(size_t)(m0 + r) * IN_FEATS + c];
  }
  __syncthreads();

  const int ml  = lane & 15;
  const int klo = (lane < 16) ? 0 : 2;

  v8f acc = {};
  #pragma unroll 4
  for (int k = 0; k < IN_FEATS; k += 4) {
    // A fragment 16x4: M = lane%16, K = {klo, klo+1} within this k-step.
    const float2 av = *(const float2*)&lA[ml * IN_FEATS + k + klo];
    v2f a; a.x = av.x; a.y = av.y;
    // B fragment 4x16: N = lane%16, same K split, row-major W [K][N].
    v2f b;
    b.x = W[(size_t)(k + klo)     * HD + n0 + ml];
    b.y = W[(size_t)(k + klo + 1) * HD + n0 + ml];
    acc = __builtin_amdgcn_wmma_f32_16x16x4_f32(
        /*neg_a=*/false, a, /*neg_b=*/false, b,
        /*c_mod=*/(short)0, acc, /*reuse_a=*/false, /*reuse_b=*/false);
  }

  // D 16x16: VGPR v -> M = v + (lane<16 ? 0 : 8), N = lane%16.
  const int rbase = m0 + ((lane < 16) ? 0 : 8);
  #pragma unroll
  for (int v = 0; v < 8; ++v) {
    h[(size_t)(rbase + v) * HD + n0 + ml] = acc[v];
  }
}

// ---------------------------------------------------------------------------
// Kernel 2: out[n, :] = bias[:]  (also clears the 0xAA poison).
// ---------------------------------------------------------------------------
__global__ void gat_init_out(float* __restrict__ out,
                             const float* __restrict__ bias) {
  const int idx = blockIdx.x * blockDim.x + threadIdx.x;
  if (idx < N_NODES * HD) out[idx] = bias[idx & (HD - 1)];
}

// ---------------------------------------------------------------------------
// Kernel 3: per-(node, head) attention logits; init m = -inf, denom = 0.
// ---------------------------------------------------------------------------
__global__ void gat_logits(const float* __restrict__ h,
                           const float* __restrict__ attn,
                           float* __restrict__ es, float* __restrict__ ed,
                           float* __restrict__ m, float* __restrict__ denom) {
  const int idx = blockIdx.x * blockDim.x + threadIdx.x;   // n*8 + head
  if (idx >= N_NODES * NUM_HEADS) return;
  const int head = idx & (NUM_HEADS - 1);
  const int n    = idx >> 3;
  const float* hp = h    + (size_t)n * HD + head * OUT_FEATS;
  const float* a0 = attn +           head * OUT_FEATS;      // source side
  const float* a1 = attn + HD      + head * OUT_FEATS;      // dest side
  float s0 = 0.f, s1 = 0.f;
  #pragma unroll
  for (int d = 0; d < OUT_FEATS; ++d) {
    const float hv = hp[d];
    s0 += hv * a0[d];
    s1 += hv * a1[d];
  }
  es[idx]    = s0;
  ed[idx]    = s1;
  m[idx]     = -INFINITY;
  denom[idx] = 0.f;
}

// ---------------------------------------------------------------------------
// Float atomic max without CAS: sign-split int/uint trick (monotone in IEEE
// bit patterns for each sign; mixed-sign case resolves correctly since
// non-negative int bits always beat negative int bits, and negative uint
// bits are always >= 0x80000000).
// ---------------------------------------------------------------------------
__device__ __forceinline__ void atomicMaxFloat(float* addr, float val) {
  if (val >= 0.f) {
    atomicMax((int*)addr, __float_as_int(val));
  } else {
    atomicMin((unsigned int*)addr, __float_as_uint(val));
  }
}

__device__ __forceinline__ float edge_score(const float* __restrict__ es,
                                            const float* __restrict__ ed,
                                            int s, int t, int head) {
  const float e = es[s * NUM_HEADS + head] + ed[t * NUM_HEADS + head];
  return e > 0.f ? e : NEG_SLOPE * e;   // LeakyReLU
}

// ---------------------------------------------------------------------------
// Kernel 4: segment max of edge scores over dst.
// ---------------------------------------------------------------------------
__global__ void gat_edge_max(const float* __restrict__ es,
                             const float* __restrict__ ed,
                             const int* __restrict__ src,
                             const int* __restrict__ dst,
                             float* __restrict__ m) {
  const int idx = blockIdx.x * blockDim.x + threadIdx.x;   // e*8 + head
  if (idx >= N_EDGES * NUM_HEADS) return;
  const int head = idx & (NUM_HEADS - 1);
  const int e    = idx >> 3;
  const int s = src[e], t = dst[e];
  atomicMaxFloat(&m[t * NUM_HEADS + head], edge_score(es, ed, s, t, head));
}

// ---------------------------------------------------------------------------
// Kernel 5: segment sum of exp(e - m[dst]) over dst.
// ---------------------------------------------------------------------------
__global__ void gat_edge_sum(const float* __restrict__ es,
                             const float* __restrict__ ed,
                             const int* __restrict__ src,
                             const int* __restrict__ dst,
                             const float* __restrict__ m,
                             float* __restrict__ denom) {
  const int idx = blockIdx.x * blockDim.x + threadIdx.x;
  if (idx >= N_EDGES * NUM_HEADS) return;
  const int head = idx & (NUM_HEADS - 1);
  const int e    = idx >> 3;
  const int s = src[e], t = dst[e];
  const float v  = edge_score(es, ed, s, t, head);
  const float ex = __expf(v - m[t * NUM_HEADS + head]);
  atomicAdd(&denom[t * NUM_HEADS + head], ex);
}

// ---------------------------------------------------------------------------
// Kernel 6: out[dst, head, :] += alpha * h[src, head, :]
// One thread per (edge, head): 16 fp32 atomic adds, h[src] gathered as 4x
// float4 (h is L2-resident: 25.6 MB << 192 MB).
// ---------------------------------------------------------------------------
__global__ void gat_aggregate(const float* __restrict__ h,
                              const float* __restrict__ es,
                              const float* __restrict__ ed,
                              const float* __restrict__ m,
                              const float* __restrict__ denom,
                              const int* __restrict__ src,
                              const int* __restrict__ dst,
                              float* __restrict__ out) {
  const int idx = blockIdx.x * blockDim.x + threadIdx.x;
  if (idx >= N_EDGES * NUM_HEADS) return;
  const int head = idx & (NUM_HEADS - 1);
  const int e    = idx >> 3;
  const int s = src[e], t = dst[e];
  const float v  = edge_score(es, ed, s, t, head);
  const float dn = denom[t * NUM_HEADS + head];
  const float alpha = __expf(v - m[t * NUM_HEADS + head]) / (dn > 0.f ? dn : 1.f);
  const float* hs = h   + (size_t)s * HD + head * OUT_FEATS;
  float*       op = out + (size_t)t * HD + head * OUT_FEATS;
  #pragma unroll
  for (int d = 0; d < OUT_FEATS; d += 4) {
    const float4 hv = *(const float4*)&hs[d];
    atomicAdd(&op[d + 0], alpha * hv.x);
    atomicAdd(&op[d + 1], alpha * hv.y);
    atomicAdd(&op[d + 2], alpha * hv.z);
    atomicAdd(&op[d + 3], alpha * hv.w);
  }
}

// ---------------------------------------------------------------------------
// Launcher. d_in order: feat, W, attn_weights, bias, src, dst.
// ws layout (floats): h[N*128] | es[N*8] | ed[N*8] | m[N*8] | denom[N*8]
// ---------------------------------------------------------------------------
extern "C" void kernel_launch(void* const* d_in, const int* in_sizes, int n_in,
                              void* d_out, int out_size, void* d_ws, size_t ws_size,
                              hipStream_t stream) {
  const float* feat = (const float*)d_in[0];
  const float* W    = (const float*)d_in[1];
  const float* attn = (const float*)d_in[2];
  const float* bias = (const float*)d_in[3];
  const int*   src  = (const int*)d_in[4];
  const int*   dst  = (const int*)d_in[5];
  float*       out  = (float*)d_out;

  float* h     = (float*)d_ws;
  float* es    = h     + (size_t)N_NODES * HD;
  float* ed    = es    + (size_t)N_NODES * NUM_HEADS;
  float* m     = ed    + (size_t)N_NODES * NUM_HEADS;
  float* denom = m     + (size_t)N_NODES * NUM_HEADS;

  const int NH = N_NODES * NUM_HEADS;   // 400,000
  const int EH = N_EDGES * NUM_HEADS;   // 6,400,000

  gat_init_out<<<(N_NODES * HD + 255) / 256, 256, 0, stream>>>(out, bias);
  gat_gemm_wmma<<<N_NODES / 16, 256, 0, stream>>>(feat, W, h);
  gat_logits<<<(NH + 255) / 256, 256, 0, stream>>>(h, attn, es, ed, m, denom);
  gat_edge_max<<<(EH + 255) / 256, 256, 0, stream>>>(es, ed, src, dst, m);
  gat_edge_sum<<<(EH + 255) / 256, 256, 0, stream>>>(es, ed, src, dst, m, denom);
  gat_aggregate<<<(EH + 255) / 256, 256, 0, stream>>>(h, es, ed, m, denom, src, dst, out);
}